// HyperbolicGraphConv_5007931867355
// MI455X (gfx1250) — compile-verified
//
#include <hip/hip_runtime.h>
#include <math.h>

// ---------------------------------------------------------------------------
// HyperbolicGraphConv for MI455X (gfx1250, wave32)
// Stages: logmap -> edge scatter (atomics) -> mean+exp/log roundtrip ->
//         WMMA f32 GEMM (+bias) -> expmap (in-place on d_out)
// ---------------------------------------------------------------------------

typedef __attribute__((ext_vector_type(2))) float v2f;
typedef __attribute__((ext_vector_type(8))) float v8f;

#define HEPS 1e-7f
#define CLIP_MAX (1.0f - 1e-5f)
#define DIM 128

__device__ __forceinline__ float wave_reduce_sum(float v) {
#pragma unroll
    for (int off = 16; off > 0; off >>= 1)
        v += __shfl_xor(v, off, 32);
    return v;
}

// -------- Kernel 1: v = logmap0(x); agg = v; count = 1 ----------------------
__global__ void hgc_logmap_init(const float* __restrict__ x,
                                const float* __restrict__ c_ptr,
                                float* __restrict__ v,
                                float* __restrict__ agg,
                                float* __restrict__ count,
                                int N) {
    int row  = (blockIdx.x * blockDim.x + threadIdx.x) >> 5;
    int lane = threadIdx.x & 31;
    if (row >= N) return;

    const float4* xr = (const float4*)(x + (size_t)row * DIM);
    float4 val = xr[lane];
    float ss = val.x * val.x + val.y * val.y + val.z * val.z + val.w * val.w;
    ss = wave_reduce_sum(ss);

    float n  = fmaxf(sqrtf(ss), HEPS);
    float sc = sqrtf(*c_ptr);
    float y  = fminf(sc * n, CLIP_MAX);
    float s  = atanhf(y) / (sc * n);

    val.x *= s; val.y *= s; val.z *= s; val.w *= s;
    ((float4*)(v   + (size_t)row * DIM))[lane] = val;
    ((float4*)(agg + (size_t)row * DIM))[lane] = val;
    if (lane == 0) count[row] = 1.0f;
}

// -------- Kernel 2: agg[dst] += v[src]; count[dst] += 1 ---------------------
__global__ void hgc_scatter(const float* __restrict__ v,
                            const int* __restrict__ src,
                            const int* __restrict__ dst,
                            float* __restrict__ agg,
                            float* __restrict__ count,
                            int E) {
    unsigned gid  = blockIdx.x * blockDim.x + threadIdx.x;
    int      edge = gid >> 5;
    int      lane = gid & 31;
    if (edge >= E) return;

    int s = src[edge];
    int d = dst[edge];

    float4 val = ((const float4*)(v + (size_t)s * DIM))[lane];
    float* base = agg + (size_t)d * DIM + lane * 4;
    unsafeAtomicAdd(base + 0, val.x);
    unsafeAtomicAdd(base + 1, val.y);
    unsafeAtomicAdd(base + 2, val.z);
    unsafeAtomicAdd(base + 3, val.w);
    if (lane == 0) unsafeAtomicAdd(count + d, 1.0f);
}

// -------- Kernel 3: v2 = logmap0(expmap0(agg / count))  (in-place) ----------
__global__ void hgc_combine(float* __restrict__ agg,
                            const float* __restrict__ count,
                            const float* __restrict__ c_ptr,
                            int N) {
    int row  = (blockIdx.x * blockDim.x + threadIdx.x) >> 5;
    int lane = threadIdx.x & 31;
    if (row >= N) return;

    float4 a = ((float4*)(agg + (size_t)row * DIM))[lane];
    float inv = 1.0f / fmaxf(count[row], 1.0f);
    a.x *= inv; a.y *= inv; a.z *= inv; a.w *= inv;

    float ss = wave_reduce_sum(a.x * a.x + a.y * a.y + a.z * a.z + a.w * a.w);
    float sc = sqrtf(*c_ptr);

    // expmap0
    float n1 = fmaxf(sqrtf(ss), HEPS);
    float r1 = sc * n1;
    float s1 = tanhf(r1) / r1;
    // logmap0 of the result; ||x_hyp|| = s1 * n1 (s1 > 0)
    float n2 = fmaxf(s1 * n1, HEPS);
    float y  = fminf(sc * n2, CLIP_MAX);
    float s2 = atanhf(y) / (sc * n2);

    float t = s1 * s2;
    a.x *= t; a.y *= t; a.z *= t; a.w *= t;
    ((float4*)(agg + (size_t)row * DIM))[lane] = a;
}

// -------- Kernel 4: out = v2 @ W^T + bias  (V_WMMA_F32_16X16X4_F32) ---------
// Block = 256 threads = 8 waves. Each block: 16 output rows; wave w: 16 cols.
// 32-bit A (16x4) layout: lanes 0-15 -> K pair {0,1}, lanes 16-31 -> {2,3}.
// B (4x16) mirrors it: per-lane column, same K-pair split.
__global__ void hgc_gemm_wmma(const float* __restrict__ v2,
                              const float* __restrict__ W,   // (OUT, IN) row-major
                              const float* __restrict__ bias,
                              float* __restrict__ out,
                              int N) {
    int m0   = blockIdx.x * 16;
    int wave = threadIdx.x >> 5;
    int lane = threadIdx.x & 31;
    int n0   = wave * 16;
    int lm   = lane & 15;
    int koff = (lane >> 4) << 1;   // 0 for lanes 0-15, 2 for lanes 16-31

    int m = m0 + lm;
    if (m >= N) m = N - 1;         // clamp (N is a multiple of 16 anyway)

    const float* arow = v2 + (size_t)m * DIM;
    const float* brow = W + (size_t)(n0 + lm) * DIM;

    float bv = bias[n0 + lm];
    v8f acc;
#pragma unroll
    for (int j = 0; j < 8; ++j) acc[j] = bv;

#pragma unroll
    for (int k0 = 0; k0 < DIM; k0 += 4) {
        v2f a, b;
        a.x = arow[k0 + koff];
        a.y = arow[k0 + koff + 1];
        b.x = brow[k0 + koff];
        b.y = brow[k0 + koff + 1];
        acc = __builtin_amdgcn_wmma_f32_16x16x4_f32(
            /*neg_a=*/false, a, /*neg_b=*/false, b,
            /*c_mod=*/(short)0, acc, /*reuse_a=*/false, /*reuse_b=*/false);
    }

    int rbase = m0 + ((lane >> 4) << 3);  // lanes 16-31 hold rows M=8..15
#pragma unroll
    for (int j = 0; j < 8; ++j) {
        int r = rbase + j;
        if (r < N) out[(size_t)r * DIM + n0 + lm] = acc[j];
    }
}

// -------- Kernel 5: out = expmap0(out)  (in-place, wave per row) ------------
__global__ void hgc_expmap(float* __restrict__ out,
                           const float* __restrict__ c_ptr,
                           int N) {
    int row  = (blockIdx.x * blockDim.x + threadIdx.x) >> 5;
    int lane = threadIdx.x & 31;
    if (row >= N) return;

    float4 o = ((float4*)(out + (size_t)row * DIM))[lane];
    float ss = wave_reduce_sum(o.x * o.x + o.y * o.y + o.z * o.z + o.w * o.w);
    float sc = sqrtf(*c_ptr);
    float n  = fmaxf(sqrtf(ss), HEPS);
    float r  = sc * n;
    float s  = tanhf(r) / r;
    o.x *= s; o.y *= s; o.z *= s; o.w *= s;
    ((float4*)(out + (size_t)row * DIM))[lane] = o;
}

// ---------------------------------------------------------------------------
extern "C" void kernel_launch(void* const* d_in, const int* in_sizes, int n_in,
                              void* d_out, int out_size, void* d_ws, size_t ws_size,
                              hipStream_t stream) {
    const float* x      = (const float*)d_in[0];
    const int*   edges  = (const int*)d_in[1];   // [2, E] flattened: src then dst
    const float* c_ptr  = (const float*)d_in[2];
    // d_in[3] = num_nodes (derivable from in_sizes[0])
    const float* weight = (const float*)d_in[4]; // (OUT_DIM, IN_DIM)
    const float* bias   = (const float*)d_in[5];

    const int N = in_sizes[0] / DIM;
    const int E = in_sizes[1] / 2;
    const int* src = edges;
    const int* dst = edges + E;

    float* ws    = (float*)d_ws;
    float* v     = ws;                         // N * 128
    float* agg   = ws + (size_t)N * DIM;       // N * 128 (reused as v2)
    float* count = ws + (size_t)2 * N * DIM;   // N

    float* out = (float*)d_out;

    const int TB = 256;                 // 8 waves per block
    const int rows_per_block  = TB / 32;
    const int row_blocks      = (N + rows_per_block - 1) / rows_per_block;
    const int edge_blocks     = (int)(((long long)E * 32 + TB - 1) / TB);
    const int gemm_blocks     = (N + 15) / 16;

    hgc_logmap_init<<<row_blocks, TB, 0, stream>>>(x, c_ptr, v, agg, count, N);
    hgc_scatter<<<edge_blocks, TB, 0, stream>>>(v, src, dst, agg, count, E);
    hgc_combine<<<row_blocks, TB, 0, stream>>>(agg, count, c_ptr, N);
    hgc_gemm_wmma<<<gemm_blocks, TB, 0, stream>>>(agg, weight, bias, out, N);
    hgc_expmap<<<row_blocks, TB, 0, stream>>>(out, c_ptr, N);
}